// HungarianMatcher_19164144074937
// MI455X (gfx1250) — compile-verified
//
#include <hip/hip_runtime.h>
#include <hip/hip_bf16.h>

// HungarianMatcher cost matrix for MI455X (gfx1250, wave32).
// C[n,m] = 5*L1(cxcywh) + 2*(pos-neg)[n, cls[m]] - 2*GIoU(xyxy)
// N = 16*900 = 14400 predictions, M = 1600 GT, 91 classes (K padded to 92).
//
// Store-bandwidth-bound (92 MB out, ~4us floor @ 23.3 TB/s). Class term is a
// one-hot GEMM Delta(Nx92) x OneHot(92xM) on the matrix pipe
// (V_WMMA_F32_16X16X4_F32); its D layout (lane=m, vgpr=n) is reused directly
// by the VALU epilogue adding L1 + GIoU. Delta tile is staged into LDS with
// GLOBAL_LOAD_ASYNC_TO_LDS_B128 (ASYNCcnt) overlapping the box setup, and
// the streaming output uses non-temporal stores.

#define NQTOT  14400      // 16*900
#define MGT    1600
#define NCLS   91
#define KPAD   92         // 91 padded to multiple of 4

typedef __attribute__((ext_vector_type(2))) float v2f;
typedef __attribute__((ext_vector_type(8))) float v8f;

// ---------------- Kernel 1: focal class-cost table ----------------
// delta[n][c] = COEF_CLASS * (pos - neg), c in [0,91), col 91 zero-padded.
__global__ void hm_prep_delta(const float* __restrict__ logits,
                              float* __restrict__ delta) {
    int i = blockIdx.x * 256 + threadIdx.x;
    if (i >= NQTOT * KPAD) return;
    int n = i / KPAD;
    int c = i - n * KPAD;
    float d = 0.0f;
    if (c < NCLS) {
        float x   = logits[n * NCLS + c];
        float p   = __builtin_amdgcn_rcpf(1.0f + __expf(-x));   // sigmoid
        float omp = 1.0f - p;
        // GAMMA = 2 -> plain squares. ALPHA = 0.25.
        float pos = 0.25f * omp * omp * (-__logf(p   + 1e-8f));
        float neg = 0.75f * p   * p   * (-__logf(omp + 1e-8f)); // log1p(-p+eps)=log(1-p+eps)
        d = 2.0f * (pos - neg);                                  // fold COEF_CLASS
    }
    delta[i] = d;
}

// ---------------- Kernel 2: fused cost matrix ----------------
// Block = 256 threads = 8 waves. Block tile: 32 n x 64 m.
// Wave w: nSub = w>>2 (0..1), mSub = w&3 (0..3) -> one 16x16 WMMA tile.
// Grid: (14400/32, 1600/64) = (450, 25), exact.
__global__ __launch_bounds__(256)
void hm_cost_kernel(const float* __restrict__ delta,     // [NQTOT][KPAD]
                    const float* __restrict__ pred_boxes,// [NQTOT][4] cxcywh
                    const float* __restrict__ gt_boxes,  // [MGT][4]   cxcywh
                    const int*   __restrict__ gt_class,  // [MGT]
                    float*       __restrict__ out) {     // [NQTOT][MGT]
    __shared__ float sDelta[32 * KPAD];   // 11776 B, conflict-free A reads
    __shared__ float sPred[32 * 9];       // cx,cy,w,h,x0,y0,x1,y1,area
    __shared__ float sGt  [64 * 9];
    __shared__ int   sCls [64];

    const int tid   = threadIdx.x;
    const int nBase = blockIdx.x * 32;
    const int mBase = blockIdx.y * 64;

    // ---- async DMA: delta tile global -> LDS (no VGPR round-trip) ----
    // Tile is a contiguous 11776 B (= 736 x b128) slab; GVS mode:
    // addr = SADDR(u64 base) + VADDR(u32 byte offset). VDST = LDS byte addr
    // (low 32 bits of the generic pointer, per ISA aperture mapping).
    {
        const unsigned long long gbase =
            (unsigned long long)(delta + (size_t)nBase * KPAD);
        for (int i = tid; i < (32 * KPAD) / 4; i += 256) {
            unsigned ldsAddr = (unsigned)(unsigned long long)(void*)&sDelta[i * 4];
            unsigned gOff    = (unsigned)(i * 16);
            asm volatile("global_load_async_to_lds_b128 %0, %1, %2"
                         :
                         : "v"(ldsAddr), "v"(gOff), "s"(gbase)
                         : "memory");
        }
    }
    // ---- box extension overlaps the async DMA ----
    if (tid < 64) {                                      // gt rows
        const float4 b = ((const float4*)gt_boxes)[mBase + tid];
        float x0 = b.x - 0.5f * b.z, y0 = b.y - 0.5f * b.w;
        float x1 = b.x + 0.5f * b.z, y1 = b.y + 0.5f * b.w;
        float* r = &sGt[tid * 9];
        r[0] = b.x; r[1] = b.y; r[2] = b.z; r[3] = b.w;
        r[4] = x0;  r[5] = y0;  r[6] = x1;  r[7] = y1;
        r[8] = (x1 - x0) * (y1 - y0);
        sCls[tid] = gt_class[mBase + tid];
    } else if (tid < 96) {                               // pred rows
        const int t = tid - 64;
        const float4 b = ((const float4*)pred_boxes)[nBase + t];
        float x0 = b.x - 0.5f * b.z, y0 = b.y - 0.5f * b.w;
        float x1 = b.x + 0.5f * b.z, y1 = b.y + 0.5f * b.w;
        float* r = &sPred[t * 9];
        r[0] = b.x; r[1] = b.y; r[2] = b.z; r[3] = b.w;
        r[4] = x0;  r[5] = y0;  r[6] = x1;  r[7] = y1;
        r[8] = (x1 - x0) * (y1 - y0);
    }
    asm volatile("s_wait_asynccnt 0x0" ::: "memory");    // DMA done before barrier
    __syncthreads();

    const int wave = tid >> 5;
    const int lane = tid & 31;
    const int nSub = wave >> 2;        // 0..1
    const int mSub = wave & 3;         // 0..3
    const int l16  = lane & 15;
    const int hi   = lane >> 4;        // 0/1
    const int mLoc = mSub * 16 + l16;
    const int cls  = sCls[mLoc];       // this lane's gt class (B column m = l16)

    // ---- class term: one-hot GEMM on the matrix pipe ----
    // A (16x4 f32): lane l<16 -> row M=l, K={kb,kb+1}; lane>=16 -> row M=l-16, K={kb+2,kb+3}
    // B (4x16 f32): lane l<16 -> col N=l, K={kb,kb+1}; lane>=16 -> col N=l-16, K={kb+2,kb+3}
    v8f acc = {};
    const float* aRow = &sDelta[(nSub * 16 + l16) * KPAD + hi * 2];
    for (int kb = 0; kb < KPAD; kb += 4) {
        v2f a = *(const v2f*)(aRow + kb);            // 8B-aligned LDS read
        int k0 = kb + hi * 2;
        v2f b;
        b.x = (cls == k0)     ? 1.0f : 0.0f;
        b.y = (cls == k0 + 1) ? 1.0f : 0.0f;
        acc = __builtin_amdgcn_wmma_f32_16x16x4_f32(
                  false, a, false, b, (short)0, acc, false, false);
    }

    // ---- epilogue: L1 + GIoU in the same (lane,vgpr)->(m,n) layout ----
    const float* g = &sGt[mLoc * 9];
    const float gcx = g[0], gcy = g[1], gw = g[2], gh = g[3];
    const float gx0 = g[4], gy0 = g[5], gx1 = g[6], gy1 = g[7], gA = g[8];
    const int   mGlob = mBase + mLoc;

#pragma unroll
    for (int v = 0; v < 8; ++v) {
        const int nLoc = nSub * 16 + hi * 8 + v;     // D layout: vgpr v -> row
        const float* p = &sPred[nLoc * 9];           // LDS broadcast across lanes
        // L1 on cxcywh
        float l1 = fabsf(p[0] - gcx) + fabsf(p[1] - gcy)
                 + fabsf(p[2] - gw)  + fabsf(p[3] - gh);
        // intersection / union
        float ix0 = fmaxf(p[4], gx0), iy0 = fmaxf(p[5], gy0);
        float ix1 = fminf(p[6], gx1), iy1 = fminf(p[7], gy1);
        float iw  = fmaxf(ix1 - ix0, 0.0f), ih = fmaxf(iy1 - iy0, 0.0f);
        float inter = iw * ih;
        float uni   = p[8] + gA - inter;             // > 0 always
        float iou   = inter * __builtin_amdgcn_rcpf(uni);
        // enclosing box
        float cx0 = fminf(p[4], gx0), cy0 = fminf(p[5], gy0);
        float cx1 = fmaxf(p[6], gx1), cy1 = fmaxf(p[7], gy1);
        float cA  = (cx1 - cx0) * (cy1 - cy0);       // > 0 always
        float giou = iou - (cA - uni) * __builtin_amdgcn_rcpf(cA);
        // C = 5*L1 + class(acc, already x2) - 2*GIoU
        float r = fmaf(5.0f, l1, acc[v]) - 2.0f * giou;
        // streaming output: non-temporal store (TH=NT), keep delta hot in L2
        __builtin_nontemporal_store(r, &out[(size_t)(nBase + nLoc) * MGT + mGlob]);
    }
}

extern "C" void kernel_launch(void* const* d_in, const int* in_sizes, int n_in,
                              void* d_out, int out_size, void* d_ws, size_t ws_size,
                              hipStream_t stream) {
    const float* pred_logits = (const float*)d_in[0];
    const float* pred_boxes  = (const float*)d_in[1];
    const float* gt_boxes    = (const float*)d_in[2];
    const int*   gt_class    = (const int*)d_in[3];
    float*       out         = (float*)d_out;
    float*       delta       = (float*)d_ws;         // NQTOT*KPAD floats (5.3 MB)

    (void)in_sizes; (void)n_in; (void)out_size; (void)ws_size;

    {   // focal class-cost table
        int total  = NQTOT * KPAD;
        int blocks = (total + 255) / 256;
        hm_prep_delta<<<blocks, 256, 0, stream>>>(pred_logits, delta);
    }
    {   // fused cost matrix
        dim3 grid(NQTOT / 32, MGT / 64);             // 450 x 25, exact
        hm_cost_kernel<<<grid, dim3(256), 0, stream>>>(
            delta, pred_boxes, gt_boxes, gt_class, out);
    }
}